// LSTMCell_71562745086289
// MI455X (gfx1250) — compile-verified
//
#include <hip/hip_runtime.h>
#include <hip/hip_bf16.h>

typedef float v2f __attribute__((ext_vector_type(2)));
typedef float v8f __attribute__((ext_vector_type(8)));

#define B_   128
#define U_   2048
#define L_   512
#define UK   1024          // U/K
#define UN   512           // U/N
#define KDIM 6144          // 3*U (full reduction length)

// d_out flat layout (floats): x_out[128,4096] | h_out[128,2048] | c_out[128,2048] | ctx[128,2048]
#define XOFF   0
#define HOFF   (B_ * 2 * U_)          // 524288
#define COFF   (HOFF + B_ * U_)       // 786432
#define CTXOFF (COFF + B_ * U_)       // 1048576

#define KC   32
#define APAD 36
#define BPAD 36

// -------------------------------------------------------------------------
// Kernel 1: fused GEMM (WMMA f32 16x16x4) + LSTM gates, register-double-
// buffered global->LDS staging.
// block = 256 threads (8 waves). blockIdx.y = n (0..3), blockIdx.x = u-tile.
// Wave w computes rows [16w,16w+16) for 4 gate tiles
// (cols n*2048 + g*512 + u0 + 0..15), so i/f/g/o fuse register-locally.
// -------------------------------------------------------------------------
__global__ __launch_bounds__(256) void lstm_gemm_gates(
    const float* __restrict__ x, const float* __restrict__ h,
    const float* __restrict__ c, const float* __restrict__ context,
    const float* __restrict__ W, float* __restrict__ out)
{
  __shared__ float Al[128 * APAD];   // A tile [row][k], padded
  __shared__ float Bl[64 * BPAD];    // B tile [colLocal][k], padded (transposed)

  const int n   = blockIdx.y;
  const int u0  = blockIdx.x * 16;
  const int t   = threadIdx.x;
  const int w   = t >> 5;
  const int l   = t & 31;
  const int lhi = l >> 4;            // 0: lanes 0-15, 1: lanes 16-31
  const int llo = l & 15;

  // staging-thread geometry (fixed per thread)
  const int aI4 = (t & 7) * 4;       // A: k offset 0..28
  const int aR0 = t >> 3;            // A: row group 0..31
  const int bJ4 = (t & 15) * 4;      // B: local col 0..60
  const int bI0 = t >> 4;            // B: k row 0..15
  const int bG  = bJ4 >> 4;          // gate group of this thread's cols
  const int bJJ = bJ4 & 15;

  v8f acc[4] = {};                   // i, f, g, o accumulator tiles
  float4 aReg[4];
  float4 bReg[2];

  // ---- chunk loader: global -> registers (next chunk), plus W prefetch ----
  auto load_chunk = [&](int kc) {
    const int   kblk  = kc / 3072;
    const int   dbase = kc - kblk * 3072;          // multiple of 32
    const int   seg   = dbase >> 10;               // 0:h 1:x 2:context
    const float* src  = (seg == 0) ? h : (seg == 1 ? x : context);
    const int   abase = kblk * UK + (dbase & 1023);
    #pragma unroll
    for (int rep = 0; rep < 4; ++rep) {
      const int row = aR0 + rep * 32;
      aReg[rep] = *reinterpret_cast<const float4*>(&src[(size_t)row * U_ + abase + aI4]);
    }
    const size_t wrow0 = ((size_t)(kblk * 4 + n)) * 3072 + dbase;
    #pragma unroll
    for (int rep = 0; rep < 2; ++rep) {
      const int i = bI0 + rep * 16;
      bReg[rep] = *reinterpret_cast<const float4*>(
          &W[(wrow0 + i) * 2048 + (size_t)bG * 512 + u0 + bJJ]);
    }
    // prefetch the W stream two chunks ahead into cache (global_prefetch_b8)
    const int kpf = kc + 2 * KC;
    if (kpf < KDIM) {
      const int   kblkp  = kpf / 3072;
      const int   dbasep = kpf - kblkp * 3072;
      const size_t wrowp = ((size_t)(kblkp * 4 + n)) * 3072 + dbasep;
      __builtin_prefetch(&W[(wrowp + bI0) * 2048 + (size_t)bG * 512 + u0 + bJJ], 0, 3);
    }
  };

  // ---- registers -> LDS ----
  auto store_chunk = [&]() {
    #pragma unroll
    for (int rep = 0; rep < 4; ++rep) {
      const int row = aR0 + rep * 32;
      Al[row * APAD + aI4 + 0] = aReg[rep].x;
      Al[row * APAD + aI4 + 1] = aReg[rep].y;
      Al[row * APAD + aI4 + 2] = aReg[rep].z;
      Al[row * APAD + aI4 + 3] = aReg[rep].w;
    }
    #pragma unroll
    for (int rep = 0; rep < 2; ++rep) {
      const int i = bI0 + rep * 16;
      Bl[(bJ4 + 0) * BPAD + i] = bReg[rep].x;
      Bl[(bJ4 + 1) * BPAD + i] = bReg[rep].y;
      Bl[(bJ4 + 2) * BPAD + i] = bReg[rep].z;
      Bl[(bJ4 + 3) * BPAD + i] = bReg[rep].w;
    }
  };

  // prologue: stage chunk 0
  load_chunk(0);
  store_chunk();
  __syncthreads();

  const int arow = w * 16 + llo;
  for (int kc = 0; kc < KDIM; kc += KC) {
    const bool has_next = (kc + KC) < KDIM;
    if (has_next) load_chunk(kc + KC);   // loads in flight during WMMA phase

    // ---- WMMA: 8 K-steps of 4, 4 gate tiles each ----
    #pragma unroll
    for (int s = 0; s < 8; ++s) {
      const int kA = s * 4 + lhi * 2;            // lanes 16-31 take K=2,3
      float2 af = *reinterpret_cast<const float2*>(&Al[arow * APAD + kA]);
      v2f a; a.x = af.x; a.y = af.y;
      #pragma unroll
      for (int g = 0; g < 4; ++g) {
        float2 bf = *reinterpret_cast<const float2*>(&Bl[(g * 16 + llo) * BPAD + kA]);
        v2f b; b.x = bf.x; b.y = bf.y;
        acc[g] = __builtin_amdgcn_wmma_f32_16x16x4_f32(
            false, a, false, b, (short)0, acc[g], false, false);
      }
    }

    if (has_next) {
      __syncthreads();
      store_chunk();
      __syncthreads();
    }
  }

  // ---- fused gates: C layout VGPR r -> M = r + 8*lhi, N = llo ----
  const int u = u0 + llo;
  #pragma unroll
  for (int r = 0; r < 8; ++r) {
    const int brow = w * 16 + r + lhi * 8;
    const int idx  = brow * U_ + n * UN + u;     // [B, U] index (U = n*512 + u)
    const float iv = acc[0][r], fv = acc[1][r], gv = acc[2][r], ov = acc[3][r];
    const float is = 1.0f / (1.0f + __expf(-iv));
    const float fs = 1.0f / (1.0f + __expf(-fv));
    const float os = 1.0f / (1.0f + __expf(-ov));
    const float gt = tanhf(gv);
    const float nc = fs * c[idx] + is * gt;
    const float nh = os * tanhf(nc);
    out[HOFF + idx] = nh;
    out[COFF + idx] = nc;
    out[XOFF + brow * (2 * U_) + n * UN + u] = nh;   // x_out first half = h_out
  }
}

// -------------------------------------------------------------------------
// Kernel 2: x_out[:, U:2U] = context
// -------------------------------------------------------------------------
__global__ __launch_bounds__(256) void copy_ctx_to_xout(
    const float* __restrict__ context, float* __restrict__ out)
{
  const int idx = blockIdx.x * 256 + threadIdx.x;    // < B*U
  const int b = idx >> 11, j = idx & 2047;
  out[XOFF + b * (2 * U_) + U_ + j] = context[idx];
}

// -------------------------------------------------------------------------
// Kernel 3: scores s[b,l] = dot(enc_attn[b,l,:], h_out[b,:]).  1 wave per l.
// grid = (64, 128), block = 256 (8 waves -> 8 l's per block).
// -------------------------------------------------------------------------
__global__ __launch_bounds__(256) void attn_score(
    const float* __restrict__ enc_attn, const float* __restrict__ out,
    float* __restrict__ sws)
{
  __shared__ float hb[U_];
  const int b = blockIdx.y;
  const int t = threadIdx.x;
  for (int i = t; i < U_; i += 256) hb[i] = out[HOFF + b * U_ + i];
  __syncthreads();

  const int wv = t >> 5, lane = t & 31;
  const int l = blockIdx.x * 8 + wv;
  const float* ea = enc_attn + ((size_t)b * L_ + l) * U_;
  float sum = 0.0f;
  #pragma unroll
  for (int it = 0; it < 16; ++it) {
    const int idx = it * 128 + lane * 4;
    float4 v = *reinterpret_cast<const float4*>(&ea[idx]);
    sum += v.x * hb[idx] + v.y * hb[idx + 1] + v.z * hb[idx + 2] + v.w * hb[idx + 3];
  }
  #pragma unroll
  for (int off = 16; off > 0; off >>= 1) sum += __shfl_down(sum, off, 32);
  if (lane == 0) sws[b * L_ + l] = sum;
}

// -------------------------------------------------------------------------
// Kernel 4: softmax over L=512 per batch row, in place in ws.
// -------------------------------------------------------------------------
__global__ __launch_bounds__(256) void softmax512(float* __restrict__ sws)
{
  __shared__ float red[256];
  const int b = blockIdx.x, t = threadIdx.x;
  float* s = sws + b * L_;
  const float s0 = s[t], s1 = s[t + 256];

  red[t] = fmaxf(s0, s1);
  __syncthreads();
  for (int off = 128; off > 0; off >>= 1) {
    if (t < off) red[t] = fmaxf(red[t], red[t + off]);
    __syncthreads();
  }
  const float m = red[0];
  __syncthreads();

  const float e0 = __expf(s0 - m), e1 = __expf(s1 - m);
  red[t] = e0 + e1;
  __syncthreads();
  for (int off = 128; off > 0; off >>= 1) {
    if (t < off) red[t] += red[t + off];
    __syncthreads();
  }
  const float inv = 1.0f / red[0];
  s[t]       = e0 * inv;
  s[t + 256] = e1 * inv;
}

// -------------------------------------------------------------------------
// Kernel 5: ctx[b,u] = sum_l p[b,l] * enc_out[b,l,u].
// grid = (2, 128), block = 256; each thread owns 4 consecutive u.
// l-loop unrolled x8 so 8 independent b128 loads are in flight per thread.
// -------------------------------------------------------------------------
__global__ __launch_bounds__(256) void attn_ctx(
    const float* __restrict__ enc_out, const float* __restrict__ sws,
    float* __restrict__ out)
{
  __shared__ float p[L_];
  const int b  = blockIdx.y;
  const int t  = threadIdx.x;
  const int ub = blockIdx.x * 1024 + t * 4;
  p[t]       = sws[b * L_ + t];
  p[t + 256] = sws[b * L_ + t + 256];
  __syncthreads();

  float4 acc = make_float4(0.f, 0.f, 0.f, 0.f);
  const float* eo = enc_out + (size_t)b * L_ * U_ + ub;
  for (int l0 = 0; l0 < L_; l0 += 8) {
    float4 v[8];
    #pragma unroll
    for (int j = 0; j < 8; ++j)
      v[j] = *reinterpret_cast<const float4*>(&eo[(size_t)(l0 + j) * U_]);
    #pragma unroll
    for (int j = 0; j < 8; ++j) {
      const float pl = p[l0 + j];
      acc.x += pl * v[j].x; acc.y += pl * v[j].y;
      acc.z += pl * v[j].z; acc.w += pl * v[j].w;
    }
  }
  *reinterpret_cast<float4*>(&out[CTXOFF + b * U_ + ub]) = acc;
}

// -------------------------------------------------------------------------
extern "C" void kernel_launch(void* const* d_in, const int* in_sizes, int n_in,
                              void* d_out, int out_size, void* d_ws, size_t ws_size,
                              hipStream_t stream)
{
  (void)in_sizes; (void)n_in; (void)out_size; (void)ws_size;
  const float* x        = (const float*)d_in[0];
  const float* h        = (const float*)d_in[1];
  const float* c        = (const float*)d_in[2];
  const float* context  = (const float*)d_in[3];
  const float* W        = (const float*)d_in[4];
  const float* enc_attn = (const float*)d_in[5];
  const float* enc_out  = (const float*)d_in[6];
  float* out = (float*)d_out;
  float* sws = (float*)d_ws;          // 128*512 floats = 256 KB of scratch

  lstm_gemm_gates<<<dim3(32, 4), 256, 0, stream>>>(x, h, c, context, W, out);
  copy_ctx_to_xout<<<(B_ * U_) / 256, 256, 0, stream>>>(context, out);
  attn_score<<<dim3(64, 128), 256, 0, stream>>>(enc_attn, out, sws);
  softmax512<<<128, 256, 0, stream>>>(sws);
  attn_ctx<<<dim3(2, 128), 256, 0, stream>>>(enc_out, sws, out);
}